// RiemannianSGNNLayer_26654567039544
// MI455X (gfx1250) — compile-verified
//
#include <hip/hip_runtime.h>

#define T_STEPS 4
#define N_NODES 50000
#define DDIM    32
#define N_EDGES 800000
#define VTH     1.0f

typedef __attribute__((ext_vector_type(2))) float v2f;
typedef __attribute__((ext_vector_type(8))) float v8f;

__device__ __forceinline__ void atomAddF32(float* p, float v) {
    // relaxed, device-scope fp32 atomic -> global_atomic_add_f32 on gfx1250
    __hip_atomic_fetch_add(p, v, __ATOMIC_RELAXED, __HIP_MEMORY_SCOPE_AGENT);
}

// ---------------- kernel 1: zero agg (in d_out) and deg ----------------
__global__ void k_zero(float* __restrict__ agg, float* __restrict__ deg) {
    int i = blockIdx.x * blockDim.x + threadIdx.x;
    if (i < T_STEPS * N_NODES * DDIM) agg[i] = 0.0f;
    if (i < N_NODES) deg[i] = 0.0f;
}

// ---------------- kernel 2: degree count over dst ----------------------
__global__ void k_deg(const int* __restrict__ dst, float* __restrict__ deg) {
    int e = blockIdx.x * blockDim.x + threadIdx.x;
    if (e < N_EDGES) atomAddF32(&deg[dst[e]], 1.0f);
}

// ---------------- kernel 3: dinv = rsqrt(deg + 1) ----------------------
__global__ void k_dinv(float* __restrict__ deg) {
    int n = blockIdx.x * blockDim.x + threadIdx.x;
    if (n < N_NODES) deg[n] = rsqrtf(deg[n] + 1.0f);
}

// ---------------- kernel 4: xw = s_seq @ W via V_WMMA_F32_16X16X4_F32 --
// One wave computes a 16-row x 32-col tile of xw.  M = T*N = 200000 rows,
// 12500 tiles, 4 waves per 128-thread block -> 3125 blocks, no remainder,
// so EXEC is all-ones for every WMMA (required by the ISA).
__global__ __launch_bounds__(128) void k_gemm(const float* __restrict__ A,
                                              const float* __restrict__ W,
                                              float* __restrict__ XW) {
    const int wave = blockIdx.x * (blockDim.x >> 5) + (threadIdx.x >> 5);
    const int lane = threadIdx.x & 31;
    const int half = lane >> 4;      // 0: lanes 0-15, 1: lanes 16-31
    const int l16  = lane & 15;
    const int rowBase = wave * 16;

    // A fragments: 16x4 f32 tile per step; lane half selects K pair.
    const float* arow = A + (size_t)(rowBase + l16) * DDIM;
    v2f a[8];
#pragma unroll
    for (int k = 0; k < 8; ++k) {
        const float* p = arow + 4 * k + 2 * half;
        v2f t; t.x = p[0]; t.y = p[1];
        a[k] = t;
    }
    // B fragments: 4x16 f32 tiles of W for column halves [0,16) and [16,32).
    v2f b0[8], b1[8];
#pragma unroll
    for (int k = 0; k < 8; ++k) {
        int krow = 4 * k + 2 * half;
        v2f t0, t1;
        t0.x = W[krow * DDIM + l16];
        t0.y = W[(krow + 1) * DDIM + l16];
        t1.x = W[krow * DDIM + 16 + l16];
        t1.y = W[(krow + 1) * DDIM + 16 + l16];
        b0[k] = t0; b1[k] = t1;
    }

    v8f acc0 = {}; v8f acc1 = {};
#pragma unroll
    for (int k = 0; k < 8; ++k) {
        acc0 = __builtin_amdgcn_wmma_f32_16x16x4_f32(
            false, a[k], false, b0[k], (short)0, acc0, false, false);
        acc1 = __builtin_amdgcn_wmma_f32_16x16x4_f32(
            false, a[k], false, b1[k], (short)0, acc1, false, false);
    }

    // D layout: VGPR v -> row v (lanes 0-15) / row v+8 (lanes 16-31), col = lane%16.
    float* orow = XW + (size_t)rowBase * DDIM;
#pragma unroll
    for (int v = 0; v < 8; ++v) {
        int r = v + 8 * half;
        orow[(size_t)r * DDIM + l16]      = acc0[v];
        orow[(size_t)r * DDIM + 16 + l16] = acc1[v];
    }
}

// ---------------- kernel 5: edge scatter-aggregate ---------------------
// thread = (edge, d); 4 fp32 atomics per thread.  xw (25.6MB) lives in L2.
__global__ void k_scatter(const int* __restrict__ src, const int* __restrict__ dst,
                          const float* __restrict__ dinv, const float* __restrict__ xw,
                          float* __restrict__ agg) {
    int idx = blockIdx.x * blockDim.x + threadIdx.x;
    if (idx >= N_EDGES * DDIM) return;
    int e = idx >> 5;
    int d = idx & 31;
    int s = src[e];
    int t = dst[e];
    float norm = dinv[s] * dinv[t];
    const size_t ND = (size_t)N_NODES * DDIM;
    size_t so = (size_t)s * DDIM + d;
    size_t to = (size_t)t * DDIM + d;
#pragma unroll
    for (int tt = 0; tt < T_STEPS; ++tt) {
        atomAddF32(&agg[tt * ND + to], xw[tt * ND + so] * norm);
    }
}

// ---------------- kernel 6: self-loop + IF scan + outputs --------------
__global__ void k_final(const float* __restrict__ z, const float* __restrict__ dinv,
                        const float* __restrict__ xw, float* __restrict__ out) {
    int idx = blockIdx.x * blockDim.x + threadIdx.x;
    if (idx >= N_NODES * DDIM) return;
    int n = idx >> 5;
    float di  = dinv[n];
    float di2 = di * di;
    const size_t ND = (size_t)N_NODES * DDIM;
    float v = 0.0f, sum = 0.0f;
#pragma unroll
    for (int t = 0; t < T_STEPS; ++t) {
        float x = out[t * ND + idx] + xw[t * ND + idx] * di2;  // agg + self-loop
        sum += x;
        v += x;
        float o = (v >= VTH) ? 1.0f : 0.0f;
        v -= o * VTH;                                          // soft reset
        out[t * ND + idx] = o;                                 // o_seq (in place)
    }
    out[(size_t)T_STEPS * ND + idx] = z[idx] + sum * (1.0f / T_STEPS);  // z_new
}

extern "C" void kernel_launch(void* const* d_in, const int* in_sizes, int n_in,
                              void* d_out, int out_size, void* d_ws, size_t ws_size,
                              hipStream_t stream) {
    const float* s_seq = (const float*)d_in[0];   // [T,N,D]
    const float* z_seq = (const float*)d_in[1];   // [N,D]
    const float* W     = (const float*)d_in[2];   // [D,D]
    const int*   ei    = (const int*)d_in[3];     // [2,E]
    const int* src = ei;
    const int* dst = ei + N_EDGES;

    float* out = (float*)d_out;                   // [T*N*D] o_seq ++ [N*D] z_new
    float* deg = (float*)d_ws;                    // N floats (deg -> dinv in place)
    float* xw  = deg + 50176;                     // T*N*D floats, 16B-aligned offset

    const int TND = T_STEPS * N_NODES * DDIM;     // 6,400,000

    k_zero<<<dim3((TND + 255) / 256), dim3(256), 0, stream>>>(out, deg);
    k_deg<<<dim3((N_EDGES + 255) / 256), dim3(256), 0, stream>>>(dst, deg);
    k_dinv<<<dim3((N_NODES + 255) / 256), dim3(256), 0, stream>>>(deg);
    // 200000 rows / 16 = 12500 waves; 4 waves per block -> 3125 blocks exactly.
    k_gemm<<<dim3(3125), dim3(128), 0, stream>>>(s_seq, W, xw);
    k_scatter<<<dim3((N_EDGES * DDIM + 255) / 256), dim3(256), 0, stream>>>(
        src, dst, deg, xw, out);
    k_final<<<dim3((N_NODES * DDIM + 255) / 256), dim3(256), 0, stream>>>(
        z_seq, deg, xw, out);
}